// FILConv_10264971837830
// MI455X (gfx1250) — compile-verified
//
#include <hip/hip_runtime.h>
#include <hip/hip_bf16.h>

#define N_NODES 50000
#define N_EDGES 800000
#define D 128   // D_IN == D_OUT == 128

typedef __attribute__((ext_vector_type(2))) float v2f;
typedef __attribute__((ext_vector_type(8))) float v8f;

// ---------------------------------------------------------------------------
// Kernel 1: zero neigh_sum [N*D] and deg [N] in workspace (must be re-zeroed
// every call; harness does not re-initialize ws between replays).
// ---------------------------------------------------------------------------
__global__ __launch_bounds__(256) void zero_ws_kernel(float* __restrict__ p, int n) {
    int i = blockIdx.x * 256 + threadIdx.x;
    int stride = gridDim.x * 256;
    for (; i < n; i += stride) p[i] = 0.0f;
}

// ---------------------------------------------------------------------------
// Kernel 2: dual GEMM via V_WMMA_F32_16X16X4_F32 (wave32, full fp32).
//   h    = feat @ W_neigh^T          -> ws
//   self = feat @ W_self^T + b_self  -> d_out (neigh term added later)
// One wave computes one 16x16 output tile for BOTH GEMMs (shared A loads).
// blockIdx.x = M tile (50000/16 = 3125), wave id (0..7) = N tile (8*16 = 128).
//
// Per ISA 7.12.2 layouts for 16x16x4 f32:
//   A (16x4):  lane%16 = M row, lane/16 selects K pair; VGPR0/1 = K, K+1
//   B (4x16):  lane%16 = N col, lane/16 selects K pair; VGPR0/1 = K, K+1
//   C/D:       lane%16 = N col, VGPR r = row r + (lane/16)*8
// B[k][n] = W[n][k], so both A and B tiles load with the identical
// (row = lane%16, kofs = (lane/16)*2) float2 pattern from row-major storage.
// ---------------------------------------------------------------------------
__global__ __launch_bounds__(256) void gemm_dual_kernel(
    const float* __restrict__ feat,
    const float* __restrict__ Wn,
    const float* __restrict__ Ws,
    const float* __restrict__ bias,
    float* __restrict__ h,
    float* __restrict__ selfout) {

    const int lane = threadIdx.x & 31;
    const int wave = threadIdx.x >> 5;          // 0..7  -> N tile
    const int m0   = blockIdx.x * 16;           // M tile base
    const int n0   = wave * 16;                 // N tile base

    const int r    = lane & 15;                 // row within tile
    const int kq   = (lane >> 4) * 2;           // 0 or 2: K sub-offset

    const float* Arow = feat + (size_t)(m0 + r) * D + kq;
    const float* Bn   = Wn   + (size_t)(n0 + r) * D + kq;
    const float* Bs   = Ws   + (size_t)(n0 + r) * D + kq;

    v8f accN = {};
    v8f accS = {};

    #pragma unroll 4
    for (int k = 0; k < D; k += 4) {
        v2f a  = *(const v2f*)(Arow + k);
        v2f bn = *(const v2f*)(Bn + k);
        v2f bs = *(const v2f*)(Bs + k);
        accN = __builtin_amdgcn_wmma_f32_16x16x4_f32(
            false, a, false, bn, (short)0, accN, false, false);
        accS = __builtin_amdgcn_wmma_f32_16x16x4_f32(
            false, a, false, bs, (short)0, accS, false, false);
    }

    const int col   = n0 + (lane & 15);
    const int rbase = m0 + (lane >> 4) * 8;
    const float bv  = bias[col];
    #pragma unroll
    for (int i = 0; i < 8; ++i) {
        const size_t row = (size_t)(rbase + i);
        h[row * D + col]       = accN[i];
        selfout[row * D + col] = accS[i] + bv;
    }
}

// ---------------------------------------------------------------------------
// Kernel 3: edge gather/scale/scatter. One wave32 per edge.
// Lane l handles features [4l, 4l+4): coalesced 512B gather per wave
// (global_load_b128 per lane) + 4 global_atomic_add_f32 per lane.
// h and neigh (51 MB total) are L2-resident (192 MB L2), so the random
// src/dst indexing stays on-chip.
// ---------------------------------------------------------------------------
__global__ __launch_bounds__(256) void edge_scatter_kernel(
    const long long* __restrict__ src,
    const long long* __restrict__ dst,
    const float* __restrict__ ew,
    const float* __restrict__ h,
    float* __restrict__ neigh,
    float* __restrict__ deg) {

    const int e = (blockIdx.x * 256 + threadIdx.x) >> 5;
    if (e >= N_EDGES) return;
    const int lane = threadIdx.x & 31;

    const long long s = src[e];
    const long long d = dst[e];
    const float w = ew[e];

    const float4 v = *(const float4*)(h + (size_t)s * D + lane * 4);
    float* out = neigh + (size_t)d * D + lane * 4;
    atomicAdd(out + 0, v.x * w);
    atomicAdd(out + 1, v.y * w);
    atomicAdd(out + 2, v.z * w);
    atomicAdd(out + 3, v.w * w);
    if (lane == 0) atomicAdd(deg + d, 1.0f);
}

// ---------------------------------------------------------------------------
// Kernel 4: out[n][c] += neigh[n][c] / max(deg[n], 1)
// ---------------------------------------------------------------------------
__global__ __launch_bounds__(256) void finalize_kernel(
    float* __restrict__ out,
    const float* __restrict__ neigh,
    const float* __restrict__ deg) {

    const int i = blockIdx.x * 256 + threadIdx.x;
    if (i >= N_NODES * D) return;
    const int node = i >> 7;
    const float inv = 1.0f / fmaxf(deg[node], 1.0f);
    out[i] += neigh[i] * inv;
}

// ---------------------------------------------------------------------------
extern "C" void kernel_launch(void* const* d_in, const int* in_sizes, int n_in,
                              void* d_out, int out_size, void* d_ws, size_t ws_size,
                              hipStream_t stream) {
    const float*     feat = (const float*)d_in[0];
    const long long* src  = (const long long*)d_in[1];
    const long long* dst  = (const long long*)d_in[2];
    const float*     ew   = (const float*)d_in[3];
    const float*     Wn   = (const float*)d_in[4];
    const float*     Ws   = (const float*)d_in[5];
    const float*     bs   = (const float*)d_in[6];
    float* out = (float*)d_out;

    // Workspace layout
    float* h     = (float*)d_ws;                       // N*D floats
    float* neigh = h + (size_t)N_NODES * D;            // N*D floats
    float* deg   = neigh + (size_t)N_NODES * D;        // N floats

    // 1) zero neigh + deg (contiguous: N*D + N floats)
    {
        const int n = N_NODES * D + N_NODES;
        const int blocks = 2048;                       // grid-stride
        zero_ws_kernel<<<blocks, 256, 0, stream>>>(neigh, n);
    }

    // 2) dual GEMM: h = feat@Wn^T ; out = feat@Ws^T + b
    {
        const int mtiles = N_NODES / 16;               // 3125, exact
        gemm_dual_kernel<<<mtiles, 256, 0, stream>>>(feat, Wn, Ws, bs, h, out);
    }

    // 3) edge scatter: wave per edge
    {
        const int edges_per_block = 256 / 32;          // 8
        const int blocks = (N_EDGES + edges_per_block - 1) / edges_per_block;
        edge_scatter_kernel<<<blocks, 256, 0, stream>>>(src, dst, ew, h, neigh, deg);
    }

    // 4) finalize
    {
        const int n = N_NODES * D;
        finalize_kernel<<<(n + 255) / 256, 256, 0, stream>>>(out, neigh, deg);
    }
}